// GraphFeaturesExtractor_18665927868853
// MI455X (gfx1250) — compile-verified
//
#include <hip/hip_runtime.h>

typedef __attribute__((ext_vector_type(2))) float v2f;
typedef __attribute__((ext_vector_type(8))) float v8f;

#define LEAKY_SLOPE 0.2f

// ---------------------------------------------------------------------------
// helpers
// ---------------------------------------------------------------------------
__device__ __forceinline__ unsigned fenc(float f) {
  // order-preserving map f32 -> u32 (monotone for all finite values)
  unsigned b = __float_as_uint(f);
  return (b & 0x80000000u) ? ~b : (b | 0x80000000u);
}
__device__ __forceinline__ float fdec(unsigned u) {
  unsigned b = (u & 0x80000000u) ? (u & 0x7FFFFFFFu) : ~u;
  return __uint_as_float(b);
}
__device__ __forceinline__ void edge_sd(const int* __restrict__ ei, int E, int e,
                                        int& s, int& d) {
  if (e < E) { s = ei[e]; d = ei[E + e]; }
  else       { s = e - E; d = e - E; }       // appended self-loops
}

// ---------------------------------------------------------------------------
// fp32 WMMA GEMM: C[M,Nn] = A[M,K] * B[K,Nn] (+bias, optional relu)
// one wave32 per 16x64 strip of C (4 accumulators) -> each A fragment feeds
// 4 v_wmma_f32_16x16x4_f32, quartering A re-read traffic.
// Requires M%16==0, K%4==0, Nn%64==0 (true for all shapes here).
// ---------------------------------------------------------------------------
__global__ void gemm_wmma_f32(const float* __restrict__ A, const float* __restrict__ B,
                              const float* __restrict__ bias, float* __restrict__ C,
                              int M, int K, int Nn, int relu) {
  const int tm   = blockIdx.x;          // 16-row tile index
  const int tn4  = blockIdx.y;          // 64-col strip index
  const int lane = threadIdx.x & 31;
  const int half = lane >> 4;           // 0: K pair (0,1), 1: K pair (2,3)
  const int l    = lane & 15;
  const int row0 = tm * 16;
  const int col0 = tn4 * 64;

  v8f acc0 = {0.f,0.f,0.f,0.f,0.f,0.f,0.f,0.f};
  v8f acc1 = acc0, acc2 = acc0, acc3 = acc0;

  // A 16x4 f32 fragment: lanes 0-15 rows 0-15 hold K=(0,1); lanes 16-31 K=(2,3)
  const float* Ap = A + (size_t)(row0 + l) * K + 2 * half;
  // B 4x16 f32 fragment (mirror): lanes 0-15 cols hold K=(0,1); lanes 16-31 K=(2,3)
  const float* Bp = B + (size_t)(2 * half) * Nn + col0 + l;

  for (int k = 0; k < K; k += 4) {
    __builtin_prefetch(&Ap[k + 32], 0, 1);          // next A cacheline
    v2f a;
    a.x = Ap[k];
    a.y = Ap[k + 1];
    const float* br0 = Bp + (size_t)k * Nn;         // B row k (+2*half)
    const float* br1 = Bp + (size_t)(k + 1) * Nn;   // B row k+1 (+2*half)
    v2f b0, b1, b2, b3;
    b0.x = br0[0];  b0.y = br1[0];
    b1.x = br0[16]; b1.y = br1[16];
    b2.x = br0[32]; b2.y = br1[32];
    b3.x = br0[48]; b3.y = br1[48];
    acc0 = __builtin_amdgcn_wmma_f32_16x16x4_f32(false, a, false, b0, (short)0, acc0, false, false);
    acc1 = __builtin_amdgcn_wmma_f32_16x16x4_f32(false, a, false, b1, (short)0, acc1, false, false);
    acc2 = __builtin_amdgcn_wmma_f32_16x16x4_f32(false, a, false, b2, (short)0, acc2, false, false);
    acc3 = __builtin_amdgcn_wmma_f32_16x16x4_f32(false, a, false, b3, (short)0, acc3, false, false);
  }

  // C/D fragment: VGPR m -> row m (lanes 0-15) / row m+8 (lanes 16-31), col = lane&15
  v8f accs[4] = {acc0, acc1, acc2, acc3};
  for (int t = 0; t < 4; ++t) {
    int col = col0 + t * 16 + l;
    float bv = bias ? bias[col] : 0.f;
    for (int m = 0; m < 8; ++m) {
      int row = row0 + m + 8 * half;
      float v = accs[t][m] + bv;
      if (relu) v = fmaxf(v, 0.f);
      C[(size_t)row * Nn + col] = v;
    }
  }
}

// ---------------------------------------------------------------------------
// per-node attention scores: as[n,h] = sum_c xl[n,h,c]*a_src[h,c] (same for ad)
// ---------------------------------------------------------------------------
__global__ void att_scores(const float* __restrict__ xl, const float* __restrict__ a_src,
                           const float* __restrict__ a_dst, float* __restrict__ as_n,
                           float* __restrict__ ad_n, int n, int H, int Cc) {
  int t = blockIdx.x * blockDim.x + threadIdx.x;
  if (t >= n * H) return;
  int node = t / H, h = t % H;
  const float* xr  = xl + (size_t)node * H * Cc + (size_t)h * Cc;
  const float* asv = a_src + h * Cc;
  const float* adv = a_dst + h * Cc;
  float s0 = 0.f, s1 = 0.f;
  for (int c = 0; c < Cc; ++c) {
    float v = xr[c];
    s0 += v * asv[c];
    s1 += v * adv[c];
  }
  as_n[t] = s0;
  ad_n[t] = s1;
}

// ---------------------------------------------------------------------------
// pass 1: segment max of leaky-relu logits over dst (ordered-uint atomicMax)
// ---------------------------------------------------------------------------
__global__ void edge_logit_max(const int* __restrict__ ei, int E, int Etot,
                               const float* __restrict__ as_n,
                               const float* __restrict__ ad_n,
                               unsigned* __restrict__ mx, int H) {
  int t = blockIdx.x * blockDim.x + threadIdx.x;
  if (t >= Etot * H) return;
  int e = t / H, h = t % H;
  int s, d;
  edge_sd(ei, E, e, s, d);
  float lg = as_n[s * H + h] + ad_n[d * H + h];
  lg = (lg > 0.f) ? lg : LEAKY_SLOPE * lg;
  atomicMax(&mx[d * H + h], fenc(lg));
}

// ---------------------------------------------------------------------------
// pass 2: p = exp(logit - max); denom[dst] += p; agg[dst,h,:] += p * xl[src,h,:]
// one wave32 per (edge, head); lanes stride the Cc channels (coalesced)
// ---------------------------------------------------------------------------
__global__ void edge_aggregate(const int* __restrict__ ei, int E, int Etot,
                               const float* __restrict__ as_n,
                               const float* __restrict__ ad_n,
                               const unsigned* __restrict__ mx,
                               const float* __restrict__ xl,
                               float* __restrict__ dnm, float* __restrict__ agg,
                               int H, int Cc) {
  int wave = (int)((blockIdx.x * (unsigned)blockDim.x + threadIdx.x) >> 5);
  int lane = threadIdx.x & 31;
  if (wave >= Etot * H) return;
  int e = wave / H, h = wave % H;
  int s, d;
  edge_sd(ei, E, e, s, d);
  float lg = as_n[s * H + h] + ad_n[d * H + h];
  lg = (lg > 0.f) ? lg : LEAKY_SLOPE * lg;
  float p = __expf(lg - fdec(mx[d * H + h]));
  if (lane == 0) atomicAdd(&dnm[d * H + h], p);
  const float* xs = xl + (size_t)s * H * Cc + (size_t)h * Cc;
  float* ag       = agg + (size_t)d * H * Cc + (size_t)h * Cc;
  for (int c = lane; c < Cc; c += 32) atomicAdd(&ag[c], p * xs[c]);
}

// ---------------------------------------------------------------------------
// finalize (in place): agg = relu(agg / (denom + 1e-16) + bias)
// ---------------------------------------------------------------------------
__global__ void gat_finalize(float* __restrict__ agg, const float* __restrict__ dnm,
                             const float* __restrict__ bias, int n, int H, int Cc) {
  int t = blockIdx.x * blockDim.x + threadIdx.x;
  int tot = n * H * Cc;
  if (t >= tot) return;
  int ch   = t % (H * Cc);
  int node = t / (H * Cc);
  int h    = ch / Cc;
  float v = agg[t] / (dnm[node * H + h] + 1e-16f) + bias[ch];
  agg[t] = fmaxf(v, 0.f);
}

// ---------------------------------------------------------------------------
// host orchestration
// ---------------------------------------------------------------------------
static inline int cdiv(int a, int b) { return (a + b - 1) / b; }

extern "C" void kernel_launch(void* const* d_in, const int* in_sizes, int n_in,
                              void* d_out, int out_size, void* d_ws, size_t ws_size,
                              hipStream_t stream) {
  const int IN = 128, H = 4, C = 64, OUT = 512;
  const int N    = in_sizes[0] / IN;     // 50000
  const int E    = in_sizes[1] / 2;      // 800000
  const int Etot = E + N;                // + self loops

  const float* x    = (const float*)d_in[0];
  const int*   ei   = (const int*)d_in[1];
  const float* W1   = (const float*)d_in[2];
  const float* as1  = (const float*)d_in[3];
  const float* ad1  = (const float*)d_in[4];
  const float* b1   = (const float*)d_in[5];
  const float* W2   = (const float*)d_in[6];
  const float* as2  = (const float*)d_in[7];
  const float* ad2  = (const float*)d_in[8];
  const float* b2   = (const float*)d_in[9];
  const float* W3   = (const float*)d_in[10];
  const float* as3  = (const float*)d_in[11];
  const float* ad3  = (const float*)d_in[12];
  const float* b3   = (const float*)d_in[13];
  const float* Wfc  = (const float*)d_in[14];
  const float* bfc  = (const float*)d_in[15];
  float*       out  = (float*)d_out;

  // workspace layout (fp32)
  float*    bufA = (float*)d_ws;                    // N*256: h_in / aggregation target
  float*    bufB = bufA + (size_t)N * (H * C);      // N*256: xl (transformed feats)
  float*    asb  = bufB + (size_t)N * (H * C);      // N*H
  float*    adb  = asb + (size_t)N * H;             // N*H
  unsigned* mxb  = (unsigned*)(adb + (size_t)N * H);// N*H (encoded max)
  float*    dnb  = (float*)(mxb + (size_t)N * H);   // N*H (denominator)

  const int TB = 256;

  // one GAT layer: hin[N,K] -> bufA[N,Hh*Cc] (relu'd), using bufB as xl scratch
  auto run_layer = [&](const float* hin, const float* W, const float* av,
                       const float* dv, const float* bias, int K, int Hh, int Cc) {
    const int Mch = Hh * Cc;
    // 1) xl = hin @ W   (WMMA, fp32; 16x64 strip per wave)
    {
      dim3 grid(N / 16, Mch / 64);
      gemm_wmma_f32<<<grid, 32, 0, stream>>>(hin, W, nullptr, bufB, N, K, Mch, 0);
    }
    // 2) per-node attention scores
    att_scores<<<cdiv(N * Hh, TB), TB, 0, stream>>>(bufB, av, dv, asb, adb, N, Hh, Cc);
    // 3) clear max / denom / aggregation target (0u == fenc-min sentinel, 0.f)
    hipMemsetAsync(mxb, 0, (size_t)N * Hh * sizeof(unsigned), stream);
    hipMemsetAsync(dnb, 0, (size_t)N * Hh * sizeof(float), stream);
    hipMemsetAsync(bufA, 0, (size_t)N * Mch * sizeof(float), stream);
    // 4) segment max of logits over dst
    edge_logit_max<<<cdiv(Etot * Hh, TB), TB, 0, stream>>>(ei, E, Etot, asb, adb, mxb, Hh);
    // 5) unnormalized softmax aggregation (wave per edge-head)
    {
      long long waves = (long long)Etot * Hh;
      long long thr   = waves * 32;
      edge_aggregate<<<(int)((thr + TB - 1) / TB), TB, 0, stream>>>(
          ei, E, Etot, asb, adb, mxb, bufB, dnb, bufA, Hh, Cc);
    }
    // 6) normalize + bias + relu (in place in bufA)
    gat_finalize<<<cdiv(N * Mch, TB), TB, 0, stream>>>(bufA, dnb, bias, N, Hh, Cc);
  };

  // layer 1: x[N,128] -> bufA[N,256]
  run_layer(x, W1, as1, ad1, b1, IN, H, C);
  // layer 2: bufA[N,256] -> bufA[N,256]  (GEMM consumes bufA into bufB before the
  //          stream-ordered memset of bufA, so the in-place ping-pong is safe)
  run_layer(bufA, W2, as2, ad2, b2, H * C, H, C);
  // layer 3: bufA[N,256] -> bufA[N,64] (1 head, mean over 1 head == identity)
  run_layer(bufA, W3, as3, ad3, b3, H * C, 1, C);
  // final fc: out = relu(bufA[N,64] @ Wfc[64,512] + bfc)
  {
    dim3 grid(N / 16, OUT / 64);
    gemm_wmma_f32<<<grid, 32, 0, stream>>>(bufA, Wfc, bfc, out, N, C, OUT, 1);
  }
}